// ContrastiveDistanceLoss_43078521979653
// MI455X (gfx1250) — compile-verified
//
#include <hip/hip_runtime.h>
#include <math.h>

typedef __attribute__((ext_vector_type(2))) float v2f;
typedef __attribute__((ext_vector_type(8))) float v8f;
typedef int vec4i __attribute__((vector_size(4 * sizeof(int))));

#define NN 8192
#define DD 1024
#define INV_T (1.0f / 0.07f)

#define BLK 128    // output macro-tile (rows == cols)
#define KC 32      // K chunk staged through LDS
#define LDSTR 36   // padded LDS row stride (36*4B = 144B: 16B aligned, conflict-free)
#define NCHUNK (DD / KC)

// Async global->LDS 16-byte copy (GLOBAL_LOAD_ASYNC_TO_LDS_B128, ASYNCcnt).
__device__ __forceinline__ void async_copy16(const float* g, float* l) {
    __builtin_amdgcn_global_load_async_to_lds_b128((vec4i*)g, (vec4i*)l, 0, 0);
}

// ---------------------------------------------------------------------------
// Per-row stats: tsq[i] = ||t_i||^2, psq[i] = ||p_i||^2, ddot[i] = t_i . p_i
// ---------------------------------------------------------------------------
__global__ __launch_bounds__(256)
void rowstats_kernel(const float* __restrict__ pred, const float* __restrict__ targ,
                     float* __restrict__ tsq, float* __restrict__ psq,
                     float* __restrict__ ddot) {
    const int i = blockIdx.x;
    const float* t = targ + (size_t)i * DD;
    const float* p = pred + (size_t)i * DD;
    float st = 0.f, sp = 0.f, sd = 0.f;
    for (int d = threadIdx.x; d < DD; d += 256) {
        float tv = t[d], pv = p[d];
        st = fmaf(tv, tv, st);
        sp = fmaf(pv, pv, sp);
        sd = fmaf(tv, pv, sd);
    }
#pragma unroll
    for (int off = 16; off > 0; off >>= 1) {
        st += __shfl_xor(st, off, 32);
        sp += __shfl_xor(sp, off, 32);
        sd += __shfl_xor(sd, off, 32);
    }
    __shared__ float r0[8], r1[8], r2[8];
    const int wave = threadIdx.x >> 5, lane = threadIdx.x & 31;
    if (lane == 0) { r0[wave] = st; r1[wave] = sp; r2[wave] = sd; }
    __syncthreads();
    if (threadIdx.x == 0) {
        float a = 0.f, b = 0.f, c = 0.f;
        for (int w = 0; w < 8; ++w) { a += r0[w]; b += r1[w]; c += r2[w]; }
        tsq[i] = a; psq[i] = b; ddot[i] = c;
    }
}

// ---------------------------------------------------------------------------
// Init reduction state (must run every call; harness does not re-poison)
// ---------------------------------------------------------------------------
__global__ void init_kernel(float* __restrict__ rowmax, float* __restrict__ colmax,
                            float* __restrict__ rowsum, float* __restrict__ colsum) {
    const int i = blockIdx.x * blockDim.x + threadIdx.x;
    if (i < NN) {
        rowmax[i] = -__builtin_inff();
        colmax[i] = -__builtin_inff();
        rowsum[i] = 0.f;
        colsum[i] = 0.f;
    }
}

// ---------------------------------------------------------------------------
// Fused GEMM tile kernel; double-buffered LDS filled by async DMA.
// PASS 1: row/col maxima of logits via atomicMax.
// PASS 2: row/col sums of exp(logit - max) via atomicAdd.
// logits[i,j] = (2 * t_i.p_j - tsq[i] - psq[j]) / 0.07
// ---------------------------------------------------------------------------
template <int PASS>
__global__ __launch_bounds__(256)
void tile_kernel(const float* __restrict__ pred, const float* __restrict__ targ,
                 const float* __restrict__ tsq, const float* __restrict__ psq,
                 float* __restrict__ rowmax, float* __restrict__ colmax,
                 float* __restrict__ rowsum, float* __restrict__ colsum) {
    __shared__ float Ts[2][BLK * LDSTR];
    __shared__ float Ps[2][BLK * LDSTR];
    __shared__ float tsqS[BLK];
    __shared__ float psqS[BLK];

    const int rowBase = blockIdx.y * BLK;  // target rows
    const int colBase = blockIdx.x * BLK;  // prediction cols
    const int tid = threadIdx.x;
    const int lane = tid & 31;
    const int wave = tid >> 5;
    const int laneLo = lane & 15;
    const int hi = lane >> 4;            // half-wave select
    const int wr = (wave >> 1) * 32;     // wave row offset in macro-tile
    const int wc = (wave & 1) * 64;      // wave col offset in macro-tile

    if (tid < BLK) {
        tsqS[tid] = tsq[rowBase + tid];
        psqS[tid] = psq[colBase + tid];
    }

    // Issue one chunk's worth of async loads into LDS buffer `buf`.
    auto stage = [&](int buf, int kc) {
#pragma unroll
        for (int it = 0; it < 4; ++it) {
            const int idx = tid + it * 256;  // 0..1023
            const int r = idx >> 3;
            const int c4 = (idx & 7) * 4;
            async_copy16(&targ[(size_t)(rowBase + r) * DD + kc + c4],
                         &Ts[buf][r * LDSTR + c4]);
            async_copy16(&pred[(size_t)(colBase + r) * DD + kc + c4],
                         &Ps[buf][r * LDSTR + c4]);
        }
    };

    v8f acc[2][4] = {};

    stage(0, 0);  // prologue: fill buffer 0

    for (int chunk = 0; chunk < NCHUNK; ++chunk) {
        const int buf = chunk & 1;
        __builtin_amdgcn_s_wait_asynccnt(0);  // own DMAs for `buf` landed
        __syncthreads();                      // everyone's DMAs landed; prev compute done
        if (chunk + 1 < NCHUNK) stage(buf ^ 1, (chunk + 1) * KC);

        const float* TsB = &Ts[buf][0];
        const float* PsB = &Ps[buf][0];
#pragma unroll
        for (int kk = 0; kk < KC; kk += 4) {
            const int ks = kk + 2 * hi;  // ISA fp32 A/B layout: half-waves hold K pairs
            v2f a[2], b[4];
#pragma unroll
            for (int tr = 0; tr < 2; ++tr)
                a[tr] = *reinterpret_cast<const v2f*>(&TsB[(wr + tr * 16 + laneLo) * LDSTR + ks]);
#pragma unroll
            for (int tc = 0; tc < 4; ++tc)
                b[tc] = *reinterpret_cast<const v2f*>(&PsB[(wc + tc * 16 + laneLo) * LDSTR + ks]);
#pragma unroll
            for (int tr = 0; tr < 2; ++tr)
#pragma unroll
                for (int tc = 0; tc < 4; ++tc)
                    acc[tr][tc] = __builtin_amdgcn_wmma_f32_16x16x4_f32(
                        false, a[tr], false, b[tc], (short)0, acc[tr][tc], false, false);
        }
    }

    // C layout: VGPR v of tile (tr,tc) holds row (v + 8*hi), col laneLo.
    auto logitf = [&](int tr, int tc, int v) -> float {
        const float c = acc[tr][tc][v];
        return (2.0f * c - tsqS[wr + tr * 16 + v + 8 * hi] - psqS[wc + tc * 16 + laneLo]) * INV_T;
    };

    if (PASS == 1) {
        // Row maxima: reduce over 4 col-tiles, then 16 lanes of the half-wave.
#pragma unroll
        for (int tr = 0; tr < 2; ++tr)
#pragma unroll
            for (int v = 0; v < 8; ++v) {
                float m = -__builtin_inff();
#pragma unroll
                for (int tc = 0; tc < 4; ++tc) m = fmaxf(m, logitf(tr, tc, v));
#pragma unroll
                for (int off = 1; off < 16; off <<= 1) m = fmaxf(m, __shfl_xor(m, off, 32));
                if (laneLo == 0)
                    atomicMax(&rowmax[rowBase + wr + tr * 16 + v + 8 * hi], m);
            }
        // Col maxima: reduce over 2 row-tiles * 8 VGPRs, then merge half-waves.
#pragma unroll
        for (int tc = 0; tc < 4; ++tc) {
            float m = -__builtin_inff();
#pragma unroll
            for (int tr = 0; tr < 2; ++tr)
#pragma unroll
                for (int v = 0; v < 8; ++v) m = fmaxf(m, logitf(tr, tc, v));
            m = fmaxf(m, __shfl_xor(m, 16, 32));
            if (hi == 0)
                atomicMax(&colmax[colBase + wc + tc * 16 + laneLo], m);
        }
    } else {
        // Row sums of exp(logit - rowmax)
#pragma unroll
        for (int tr = 0; tr < 2; ++tr)
#pragma unroll
            for (int v = 0; v < 8; ++v) {
                const float rm = rowmax[rowBase + wr + tr * 16 + v + 8 * hi];
                float s = 0.f;
#pragma unroll
                for (int tc = 0; tc < 4; ++tc) s += __expf(logitf(tr, tc, v) - rm);
#pragma unroll
                for (int off = 1; off < 16; off <<= 1) s += __shfl_xor(s, off, 32);
                if (laneLo == 0)
                    atomicAdd(&rowsum[rowBase + wr + tr * 16 + v + 8 * hi], s);
            }
        // Col sums of exp(logit - colmax)
#pragma unroll
        for (int tc = 0; tc < 4; ++tc) {
            const float cm = colmax[colBase + wc + tc * 16 + laneLo];
            float s = 0.f;
#pragma unroll
            for (int tr = 0; tr < 2; ++tr)
#pragma unroll
                for (int v = 0; v < 8; ++v) s += __expf(logitf(tr, tc, v) - cm);
            s += __shfl_xor(s, 16, 32);
            if (hi == 0)
                atomicAdd(&colsum[colBase + wc + tc * 16 + laneLo], s);
        }
    }
}

// ---------------------------------------------------------------------------
// Final scalar: 0.5 * mean( (rowLSE - diag) + (colLSE - diag) )
// ---------------------------------------------------------------------------
__global__ __launch_bounds__(256)
void finalize_kernel(const float* __restrict__ tsq, const float* __restrict__ psq,
                     const float* __restrict__ ddot,
                     const float* __restrict__ rowmax, const float* __restrict__ colmax,
                     const float* __restrict__ rowsum, const float* __restrict__ colsum,
                     float* __restrict__ out) {
    __shared__ double red[256];
    double s = 0.0;
    for (int i = threadIdx.x; i < NN; i += 256) {
        const float diag = (2.0f * ddot[i] - tsq[i] - psq[i]) * INV_T;
        const float lr = rowmax[i] + logf(rowsum[i]);
        const float lc = colmax[i] + logf(colsum[i]);
        s += (double)(lr - diag) + (double)(lc - diag);
    }
    red[threadIdx.x] = s;
    __syncthreads();
    for (int off = 128; off > 0; off >>= 1) {
        if (threadIdx.x < off) red[threadIdx.x] += red[threadIdx.x + off];
        __syncthreads();
    }
    if (threadIdx.x == 0) out[0] = (float)(red[0] * 0.5 / (double)NN);
}

// ---------------------------------------------------------------------------
extern "C" void kernel_launch(void* const* d_in, const int* in_sizes, int n_in,
                              void* d_out, int out_size, void* d_ws, size_t ws_size,
                              hipStream_t stream) {
    const float* pred = (const float*)d_in[0];  // "prediction"
    const float* targ = (const float*)d_in[1];  // "target"
    float* out = (float*)d_out;

    float* ws = (float*)d_ws;           // 7 * 8192 floats = 229 KB
    float* tsq    = ws + 0 * NN;
    float* psq    = ws + 1 * NN;
    float* ddot   = ws + 2 * NN;
    float* rowmax = ws + 3 * NN;
    float* colmax = ws + 4 * NN;
    float* rowsum = ws + 5 * NN;
    float* colsum = ws + 6 * NN;

    rowstats_kernel<<<NN, 256, 0, stream>>>(pred, targ, tsq, psq, ddot);
    init_kernel<<<(NN + 255) / 256, 256, 0, stream>>>(rowmax, colmax, rowsum, colsum);

    dim3 grid(NN / BLK, NN / BLK);
    tile_kernel<1><<<grid, 256, 0, stream>>>(pred, targ, tsq, psq,
                                             rowmax, colmax, rowsum, colsum);
    tile_kernel<2><<<grid, 256, 0, stream>>>(pred, targ, tsq, psq,
                                             rowmax, colmax, rowsum, colsum);

    finalize_kernel<<<1, 256, 0, stream>>>(tsq, psq, ddot, rowmax, colmax,
                                           rowsum, colsum, out);
}